// GraphAttentionHead_30159260352616
// MI455X (gfx1250) — compile-verified
//
#include <hip/hip_runtime.h>
#include <hip/hip_bf16.h>

#define GAT_N      50000
#define GAT_E      1600000
#define GAT_IN     128
#define GAT_OUT    64
#define GAT_SLOPE  0.2f
#define GAT_EPS    1e-12f

typedef __attribute__((ext_vector_type(2))) float v2f;
typedef __attribute__((ext_vector_type(4))) float v4f;
typedef __attribute__((ext_vector_type(8))) float v8f;

// -------------------- init: zero out, -inf max, 0 sum --------------------
__global__ void k_init(float* __restrict__ out, float* __restrict__ maxb,
                       float* __restrict__ sumb) {
  int i = blockIdx.x * blockDim.x + threadIdx.x;
  if (i < GAT_N * GAT_OUT) out[i] = 0.f;
  if (i < GAT_N) { maxb[i] = -__builtin_inff(); sumb[i] = 0.f; }
}

// -------------------- Wh = x @ W^T via V_WMMA_F32_16X16X4_F32 ------------
// One wave computes a 16x64 tile of Wh. 50000 = 3125 * 16 exactly.
__global__ __launch_bounds__(128) void k_gemm(const float* __restrict__ x,
                                              const float* __restrict__ W,
                                              float* __restrict__ Wh) {
  const int lane = threadIdx.x & 31;
  const int tile = blockIdx.x * 4 + (threadIdx.x >> 5);
  if (tile >= GAT_N / 16) return;                  // wave-uniform branch
  const int base  = tile * 16;
  const int l15   = lane & 15;
  const int khalf = (lane >> 4) << 1;              // 0 or 2 (A/B K-pair select)

  // A: row base+l15 of x;  B[k][n] = W[n*IN + k]  (W is [out,in] row-major)
  const float* xr = x + (size_t)(base + l15) * GAT_IN + khalf;
  const float* w0 = W + (size_t)(l15 +  0) * GAT_IN + khalf;
  const float* w1 = W + (size_t)(l15 + 16) * GAT_IN + khalf;
  const float* w2 = W + (size_t)(l15 + 32) * GAT_IN + khalf;
  const float* w3 = W + (size_t)(l15 + 48) * GAT_IN + khalf;

  v8f acc0 = {}, acc1 = {}, acc2 = {}, acc3 = {};

#pragma unroll 8
  for (int k = 0; k < GAT_IN; k += 4) {
    v2f a  = *(const v2f*)(xr + k);
    v2f b0 = *(const v2f*)(w0 + k);
    v2f b1 = *(const v2f*)(w1 + k);
    v2f b2 = *(const v2f*)(w2 + k);
    v2f b3 = *(const v2f*)(w3 + k);
    acc0 = __builtin_amdgcn_wmma_f32_16x16x4_f32(false, a, false, b0, (short)0, acc0, false, false);
    acc1 = __builtin_amdgcn_wmma_f32_16x16x4_f32(false, a, false, b1, (short)0, acc1, false, false);
    acc2 = __builtin_amdgcn_wmma_f32_16x16x4_f32(false, a, false, b2, (short)0, acc2, false, false);
    acc3 = __builtin_amdgcn_wmma_f32_16x16x4_f32(false, a, false, b3, (short)0, acc3, false, false);
  }

  // C/D layout: VGPR v -> row (v | v+8), col = lane&15 (+16 per N-tile)
  const int rhalf = (lane >> 4) << 3;
  float* o = Wh + (size_t)(base + rhalf) * GAT_OUT + l15;
#pragma unroll
  for (int v = 0; v < 8; ++v) {
    o[(size_t)v * GAT_OUT +  0] = acc0[v];
    o[(size_t)v * GAT_OUT + 16] = acc1[v];
    o[(size_t)v * GAT_OUT + 32] = acc2[v];
    o[(size_t)v * GAT_OUT + 48] = acc3[v];
  }
}

// -------------------- el = Wh@a_l, er = Wh@a_r ---------------------------
__global__ void k_elr(const float* __restrict__ Wh, const float* __restrict__ al,
                      const float* __restrict__ ar, float* __restrict__ el,
                      float* __restrict__ er) {
  int i = blockIdx.x * blockDim.x + threadIdx.x;
  if (i >= GAT_N) return;
  const v4f* r = (const v4f*)(Wh + (size_t)i * GAT_OUT);
  float sl = 0.f, sr = 0.f;
#pragma unroll
  for (int q = 0; q < GAT_OUT / 4; ++q) {
    v4f v = r[q];
#pragma unroll
    for (int j = 0; j < 4; ++j) {
      sl += v[j] * al[q * 4 + j];
      sr += v[j] * ar[q * 4 + j];
    }
  }
  el[i] = sl; er[i] = sr;
}

// -------------------- float atomic max via int min/max trick -------------
__device__ __forceinline__ void atomic_max_f32(float* addr, float val) {
  if (val >= 0.f) atomicMax((int*)addr, __float_as_int(val));
  else            atomicMin((unsigned int*)addr, (unsigned int)__float_as_int(val));
}

// -------------------- edge pass 1: leaky logits + segment max ------------
__global__ void k_edge_logit(const int* __restrict__ src, const int* __restrict__ dst,
                             const float* __restrict__ el, const float* __restrict__ er,
                             float* __restrict__ ebuf, float* __restrict__ maxb) {
  int e = blockIdx.x * blockDim.x + threadIdx.x;
  if (e >= GAT_E) return;
  int s = src[e], d = dst[e];
  float v = el[s] + er[d];
  v = (v > 0.f) ? v : GAT_SLOPE * v;
  ebuf[e] = v;
  atomic_max_f32(maxb + d, v);
}

// -------------------- edge pass 2: exp(e - max[dst]) + segment sum -------
__global__ void k_edge_exp(const int* __restrict__ dst, float* __restrict__ ebuf,
                           const float* __restrict__ maxb, float* __restrict__ sumb) {
  int e = blockIdx.x * blockDim.x + threadIdx.x;
  if (e >= GAT_E) return;
  int d = dst[e];
  float ex = __expf(ebuf[e] - maxb[d]);
  ebuf[e] = ex;
  atomicAdd(sumb + d, ex);
}

// -------------------- edge pass 3: out[dst] += expx * Wh[src] ------------
// Thread per (edge, feature); 64 consecutive threads share one edge, so
// Wh loads and atomics are fully coalesced; index/weight loads are uniform.
__global__ void k_scatter(const int* __restrict__ src, const int* __restrict__ dst,
                          const float* __restrict__ ebuf, const float* __restrict__ Wh,
                          float* __restrict__ out) {
  int idx = blockIdx.x * blockDim.x + threadIdx.x;
  if (idx >= GAT_E * GAT_OUT) return;
  int e = idx >> 6, f = idx & 63;
  int s = src[e], d = dst[e];
  atomicAdd(out + (size_t)d * GAT_OUT + f, ebuf[e] * Wh[(size_t)s * GAT_OUT + f]);
}

// -------------------- finalize: normalize + bias -------------------------
__global__ void k_final(float* __restrict__ out, const float* __restrict__ sumb,
                        const float* __restrict__ bias) {
  int idx = blockIdx.x * blockDim.x + threadIdx.x;
  if (idx >= GAT_N * GAT_OUT) return;
  int i = idx >> 6, f = idx & 63;
  out[idx] = out[idx] / (sumb[i] + GAT_EPS) + bias[f];
}

extern "C" void kernel_launch(void* const* d_in, const int* in_sizes, int n_in,
                              void* d_out, int out_size, void* d_ws, size_t ws_size,
                              hipStream_t stream) {
  const float* x    = (const float*)d_in[0];
  const int*   ei   = (const int*)d_in[1];
  const float* W    = (const float*)d_in[2];
  const float* al   = (const float*)d_in[3];
  const float* ar   = (const float*)d_in[4];
  const float* bias = (const float*)d_in[5];
  float* out = (float*)d_out;
  (void)in_sizes; (void)n_in; (void)out_size; (void)ws_size;

  // workspace layout (floats): Wh | el | er | maxb | sumb | ebuf  (~20 MB)
  float* ws   = (float*)d_ws;
  float* Wh   = ws;
  float* el   = Wh   + (size_t)GAT_N * GAT_OUT;
  float* er   = el   + GAT_N;
  float* maxb = er   + GAT_N;
  float* sumb = maxb + GAT_N;
  float* ebuf = sumb + GAT_N;

  const int* src = ei;
  const int* dst = ei + GAT_E;

  const int B = 256;
  k_init      <<<(GAT_N * GAT_OUT + B - 1) / B, B, 0, stream>>>(out, maxb, sumb);
  k_gemm      <<<(GAT_N / 16 + 3) / 4, 128, 0, stream>>>(x, W, Wh);
  k_elr       <<<(GAT_N + B - 1) / B, B, 0, stream>>>(Wh, al, ar, el, er);
  k_edge_logit<<<(GAT_E + B - 1) / B, B, 0, stream>>>(src, dst, el, er, ebuf, maxb);
  k_edge_exp  <<<(GAT_E + B - 1) / B, B, 0, stream>>>(dst, ebuf, maxb, sumb);
  k_scatter   <<<(GAT_E * GAT_OUT + B - 1) / B, B, 0, stream>>>(src, dst, ebuf, Wh, out);
  k_final     <<<(GAT_N * GAT_OUT + B - 1) / B, B, 0, stream>>>(out, sumb, bias);
}